// ModelNew_80908593923260
// MI455X (gfx1250) — compile-verified
//
#include <hip/hip_runtime.h>
#include <hip/hip_bf16.h>
#include <stdint.h>

// ---------------- problem constants ----------------
#define KSZ   5
#define OD    17        // valid conv output extent per dim (d,h,w in [0,17))
#define D_IN  32
#define CIN   32
#define COUT  64
#define WK    4000      // CIN * 5*5*5
#define NB    32

// workspace layout (in floats)
#define WS_WEFF 0          // 4000 floats
#define WS_BIAS 4000       // 1 float
#define WS_CONV 4096       // 32 * 4913 floats

typedef float v2f __attribute__((ext_vector_type(2)));
typedef float v8f __attribute__((ext_vector_type(8)));
typedef int   v4i __attribute__((ext_vector_type(4)));
typedef v4i __attribute__((address_space(1))) as1_v4i;   // global
typedef v4i __attribute__((address_space(3))) as3_v4i;   // LDS

#if defined(__gfx1250__) && defined(__HIP_DEVICE_COMPILE__) && \
    __has_builtin(__builtin_amdgcn_global_load_async_to_lds_b128) && \
    __has_builtin(__builtin_amdgcn_s_wait_asynccnt)
#define USE_ASYNC_LDS 1
#define WAIT_ASYNC(n) __builtin_amdgcn_s_wait_asynccnt(n)
#else
#define USE_ASYNC_LDS 0
#define WAIT_ASYNC(n)
#endif

#if defined(__gfx1250__) && defined(__HIP_DEVICE_COMPILE__) && \
    __has_builtin(__builtin_amdgcn_wmma_f32_16x16x4_f32)
#define USE_WMMA_F32 1
#else
#define USE_WMMA_F32 0
#endif

// ================= Kernel A: w_eff = sum_cout W, bias_sum =================
// Column sums of the 64 x 4000 weight matrix. WMMA: C += ones(16x4) * B(4x16),
// accumulated over 16 chunks of 4 couts -> every row of C = column sums.
__global__ __launch_bounds__(256)
void weff_kernel(const float* __restrict__ weight, const float* __restrict__ bias,
                 float* __restrict__ ws) {
    const int tid  = threadIdx.x;
#if USE_WMMA_F32
    const int wave = tid >> 5;
    const int lane = tid & 31;
    const int col16 = lane & 15;
    const int khalf = (lane >> 4) * 2;   // lanes 0-15 -> couts {0,1}; 16-31 -> {2,3}
    for (int tile = wave; tile < WK / 16; tile += 8) {   // 250 tiles, uniform per wave
        const int col = tile * 16 + col16;
        v8f c = {};
        v2f a; a.x = 1.0f; a.y = 1.0f;                   // all-ones A matrix
        for (int cg = 0; cg < 16; ++cg) {
            const int c0 = cg * 4 + khalf;
            v2f b;
            b.x = weight[(size_t)(c0    ) * WK + col];
            b.y = weight[(size_t)(c0 + 1) * WK + col];
            c = __builtin_amdgcn_wmma_f32_16x16x4_f32(false, a, false, b,
                                                      (short)0, c, false, false);
        }
        if (lane < 16) ws[WS_WEFF + col] = c[0];         // row M=0 lives in VGPR0, lanes 0-15
    }
#else
    for (int col = tid; col < WK; col += 256) {
        float s = 0.0f;
        for (int co = 0; co < COUT; ++co) s += weight[(size_t)co * WK + col];
        ws[WS_WEFF + col] = s;
    }
#endif
    if (tid == 0) {
        float s = 0.0f;
        for (int i = 0; i < COUT; ++i) s += bias[i];
        ws[WS_BIAS] = s;
    }
}

// ================= Kernel B: conv (strided cross-correlation) =================
// One block per (n, d). 320 threads; threads < 289 each own one (h,w) output.
// Input planes x[n,cin,id,:,:] are double-buffered into a zero-padded LDS tile
// via async global->LDS so the 25-tap loop needs no bounds checks.
#define PL_W 40                 // padded row stride (16B aligned)
#define PL_H 37                 // rows cover ih in [-2, 34]
#define PL_SZ (PL_W * PL_H)     // 1480 floats

__global__ __launch_bounds__(320)
void conv_kernel(const float* __restrict__ x, const float* __restrict__ ws_r,
                 float* __restrict__ conv_out) {
    __shared__ __align__(16) float buf[2][PL_SZ];
    const int tid = threadIdx.x;
    const int n = blockIdx.x / OD;
    const int d = blockIdx.x % OD;

    // zero both tiles once: halo stays zero, interior is overwritten each stage
    for (int i = tid; i < 2 * PL_SZ; i += 320) (&buf[0][0])[i] = 0.0f;
    __syncthreads();

    const float* weff = ws_r + WS_WEFF;

    // plane p = cin*5 + kd ; input z = 2d-2+kd (clamped for the load, compute skipped)
    auto issue = [&](int p, int which) {
        const int cin = p / 5, kd = p % 5;
        int id = 2 * d - 2 + kd;
        const int idc = id < 0 ? 0 : (id > 31 ? 31 : id);
        if (tid < 256) {
            const int r  = tid >> 3;        // input row 0..31
            const int c4 = (tid & 7) * 4;   // input col 0,4,...,28
            const float* g = x + ((size_t)(n * CIN + cin) * D_IN + idc) * 1024 + r * 32 + c4;
            float* l = &buf[which][(r + 2) * PL_W + 4 + c4];   // row+2, col+4 padding
#if USE_ASYNC_LDS
            __builtin_amdgcn_global_load_async_to_lds_b128((as1_v4i*)g, (as3_v4i*)l, 0, 0);
#else
            *(float4*)l = *(const float4*)g;
#endif
        }
    };

    const int h = tid / 17, w = tid % 17;   // valid for tid < 289
    float acc = 0.0f;

    issue(0, 0);
    for (int p = 0; p < CIN * 5; ++p) {
        if (p + 1 < CIN * 5) { issue(p + 1, (p + 1) & 1); WAIT_ASYNC(1); }
        else                 { WAIT_ASYNC(0); }
        __syncthreads();

        const int cin = p / 5, kd = p % 5;
        const int id = 2 * d - 2 + kd;
        if (id >= 0 && id < D_IN && tid < OD * OD) {
            const float* pl = &buf[p & 1][0];
            const float* wp = weff + cin * 125 + kd * 25;   // uniform -> scalar loads
            #pragma unroll
            for (int kh = 0; kh < KSZ; ++kh) {
                const int row = (2 * h + kh) * PL_W + 2 * w + 2;  // (ih+2)*PL_W + (iw+4)
                #pragma unroll
                for (int kw = 0; kw < KSZ; ++kw)
                    acc = fmaf(pl[row + kw], wp[kh * 5 + kw], acc);
            }
        }
        __syncthreads();
    }

    if (tid < OD * OD)
        conv_out[(size_t)n * (OD * OD * OD) + (size_t)d * (OD * OD) + h * OD + w] = acc;
}

// ================= Kernel C: fused max-pool(2,2) + sum-pool(3,3) =================
// out[n,a,b,c] = 27*bias_sum  unless a,b,c <= 2, where we add 27 clipped maxima.
__global__ __launch_bounds__(256)
void pool_kernel(const float* __restrict__ conv_in, const float* __restrict__ ws_r,
                 float* __restrict__ out) {
    __shared__ float cv[OD * OD * OD];   // 4913 floats
    const int n = blockIdx.x;
    const int tid = threadIdx.x;
    for (int i = tid; i < OD * OD * OD; i += 256)
        cv[i] = conv_in[(size_t)n * (OD * OD * OD) + i];
    __syncthreads();

    const float bias27 = 27.0f * ws_r[WS_BIAS];
    for (int t = tid; t < 1000; t += 256) {
        const int a = t / 100, b = (t / 10) % 10, c = t % 10;
        float s = bias27;
        if (a < 3 && b < 3 && c < 3) {
            for (int i = 3 * a; i < 3 * a + 3; ++i)
                for (int j = 3 * b; j < 3 * b + 3; ++j)
                    for (int k = 3 * c; k < 3 * c + 3; ++k) {
                        // window touching p1-index 8 includes a bias-only element -> max with 0
                        float m = (i == 8 || j == 8 || k == 8) ? 0.0f : -__builtin_inff();
                        const int d1 = 2 * i + 1 > 16 ? 16 : 2 * i + 1;
                        const int h1 = 2 * j + 1 > 16 ? 16 : 2 * j + 1;
                        const int w1 = 2 * k + 1 > 16 ? 16 : 2 * k + 1;
                        for (int dd = 2 * i; dd <= d1; ++dd)
                            for (int hh = 2 * j; hh <= h1; ++hh)
                                for (int ww = 2 * k; ww <= w1; ++ww)
                                    m = fmaxf(m, cv[(dd * OD + hh) * OD + ww]);
                        s += m;
                    }
        }
        out[(size_t)n * 1000 + t] = s;
    }
}

// ================= launch =================
extern "C" void kernel_launch(void* const* d_in, const int* in_sizes, int n_in,
                              void* d_out, int out_size, void* d_ws, size_t ws_size,
                              hipStream_t stream) {
    const float* x      = (const float*)d_in[0];
    const float* weight = (const float*)d_in[1];
    const float* bias   = (const float*)d_in[2];
    float* out  = (float*)d_out;
    float* ws   = (float*)d_ws;
    float* conv = ws + WS_CONV;

    hipLaunchKernelGGL(weff_kernel, dim3(1),       dim3(256), 0, stream, weight, bias, ws);
    hipLaunchKernelGGL(conv_kernel, dim3(NB * OD), dim3(320), 0, stream, x, ws, conv);
    hipLaunchKernelGGL(pool_kernel, dim3(NB),      dim3(256), 0, stream, conv, ws, out);
}